// ProposalTarget_3246995275893
// MI455X (gfx1250) — compile-verified
//
#include <hip/hip_runtime.h>
#include <hip/hip_bf16.h>
#include <stdint.h>

// ---------------------------------------------------------------------------
// ProposalTarget for MI455X (gfx1250, wave32).
//
// Stage 1 (dominant): 100k x 512 IoU max/argmax. VALU-bound (~1G VALU ops vs
// 1.6MB of HBM traffic -> 70ns at 23.3TB/s). CDNA5 play: stage the 8KB GT
// table into LDS with ONE Tensor-Data-Mover DMA per wave (tensor_load_to_lds
// + s_wait_tensorcnt) instead of 2048 per-lane vmem loads, then read it with
// conflict-free LDS broadcasts. No matmul structure -> WMMA not applicable.
// Pos/neg counts are wave-aggregated (ballot+popc, leader-lane atomic) to
// avoid serializing 200k same-address RMWs at the L2 atomic unit.
// ---------------------------------------------------------------------------

typedef unsigned int       u32;
typedef unsigned long long u64;

#define MAX_POS        32
#define TOTAL_SAMPLES  128
#define POS_THR        0.5f
#define NGT_MAX        512

#if __has_builtin(__builtin_amdgcn_tensor_load_to_lds) && __has_builtin(__builtin_amdgcn_s_wait_tensorcnt)
#define USE_TDM 1
typedef unsigned int u32x4 __attribute__((ext_vector_type(4)));
typedef int          i32x8 __attribute__((ext_vector_type(8)));
typedef int          i32x4 __attribute__((ext_vector_type(4)));
// therock-10.0 headers (amdgpu-toolchain / clang-23) => 6-arg builtin;
// ROCm 7.2 (clang-22) => 5-arg builtin. The TDM header ships only with the
// former, so use it as an arity probe.
#if __has_include(<hip/amd_detail/amd_gfx1250_TDM.h>)
#define TDM_LOAD(g0, g1, g2, g3, gz8) \
  __builtin_amdgcn_tensor_load_to_lds((g0), (g1), (g2), (g3), (gz8), 0)
#else
#define TDM_LOAD(g0, g1, g2, g3, gz8) \
  __builtin_amdgcn_tensor_load_to_lds((g0), (g1), (g2), (g3), 0)
#endif
#endif

__device__ __forceinline__ u64 umin64(u64 a, u64 b) { return a < b ? a : b; }

// Exact reproduction of jax.random.uniform(jax.random.key(42), (n,)) lane i.
// key data = [0, 42]; counts = iota(n) split in half:
//   i <  n/2 : threefry2x32(k, (i, i+n/2)) -> word0
//   i >= n/2 : threefry2x32(k, (i-n/2, i)) -> word1
// uniform = bitcast((bits>>9)|0x3f800000) - 1.0
__device__ __forceinline__ float jax_uniform_key42(int i, int n) {
  const u32 half = (u32)(n >> 1);
  u32 x0, x1; bool second;
  if ((u32)i < half) { x0 = (u32)i;        x1 = (u32)i + half; second = false; }
  else               { x0 = (u32)i - half; x1 = (u32)i;        second = true;  }
  const u32 ks0 = 0u, ks1 = 42u, ks2 = 0x1BD11BDAu ^ ks0 ^ ks1;
  x0 += ks0; x1 += ks1;
#define TF_R(r) { x0 += x1; x1 = (x1 << (r)) | (x1 >> (32 - (r))); x1 ^= x0; }
  TF_R(13) TF_R(15) TF_R(26) TF_R(6)   x0 += ks1; x1 += ks2 + 1u;
  TF_R(17) TF_R(29) TF_R(16) TF_R(24)  x0 += ks2; x1 += ks0 + 2u;
  TF_R(13) TF_R(15) TF_R(26) TF_R(6)   x0 += ks0; x1 += ks1 + 3u;
  TF_R(17) TF_R(29) TF_R(16) TF_R(24)  x0 += ks1; x1 += ks2 + 4u;
  TF_R(13) TF_R(15) TF_R(26) TF_R(6)   x0 += ks2; x1 += ks0 + 5u;
#undef TF_R
  const u32 bits = second ? x1 : x0;
  return __uint_as_float((bits >> 9) | 0x3f800000u) - 1.0f;
}

__global__ void pt_init(int* cnt) {
  cnt[0] = 0;  // total_pos
  cnt[1] = 0;  // total_neg
}

// ---------------------------------------------------------------------------
// Stage 1: per-ROI max IoU + argmax + threefry key, GT staged via TDM.
// ---------------------------------------------------------------------------
__global__ void __launch_bounds__(256)
pt_stage1(const float* __restrict__ rois, const float* __restrict__ gt,
          int n_rois, int n_gt,
          float* __restrict__ key_pos, float* __restrict__ key_neg,
          int* __restrict__ argm, int* __restrict__ cnt) {
  __shared__ float s_gt[NGT_MAX * 4];   // 8 KB, sole LDS object (offset 0)
  const int ngt = n_gt < NGT_MAX ? n_gt : NGT_MAX;
  const u32 elems = (u32)ngt * 4u;      // f32 elements to stage

#ifdef USE_TDM
  {
    // Tensor DMA descriptor (ISA ch.8): 1-row tile of `elems` 4-byte elements.
    const u64 gaddr   = (u64)(uintptr_t)gt;
    const u32 lds_off = (u32)(u64)(uintptr_t)(void*)s_gt;  // flat low 32 = LDS byte addr

    u32x4 g0;
    g0.x = 1u;                                               // count=1 (valid), user mode
    g0.y = lds_off;                                          // lds_addr
    g0.z = (u32)(gaddr & 0xFFFFFFFFull);                     // global_addr[31:0]
    g0.w = ((u32)(gaddr >> 32) & 0x01FFFFFFu) | (2u << 30);  // global_addr[56:32] | type=2

    i32x8 g1;
    g1[0] = (int)(2u << 16);                  // workgroup_mask=0, data_size=2 (4B)
    g1[1] = (int)((elems & 0xFFFFu) << 16);   // tensor_dim0[15:0] in [31:16]
    g1[2] = (int)((elems >> 16) | (1u << 16));// tensor_dim0[31:16] | tensor_dim1=1 (lo16)
    g1[3] = (int)((elems & 0xFFFFu) << 16);   // tensor_dim1 hi16=0 | tile_dim0=elems
    g1[4] = 1;                                // tile_dim1=1, tile_dim2=0
    g1[5] = (int)elems;                       // tensor_dim0_stride[31:0]
    g1[6] = 0;                                // stride hi / dim1_stride lo
    g1[7] = 0;

    i32x4 gz4 = {0, 0, 0, 0};
    i32x8 gz8 = {0, 0, 0, 0, 0, 0, 0, 0};
    // One DMA per wave (TDM ignores EXEC; descriptor is wave-uniform).
    TDM_LOAD(g0, g1, gz4, gz4, gz8);
    __builtin_amdgcn_s_wait_tensorcnt(0);     // each wave waits on its own DMA
  }
#else
  for (int t = threadIdx.x; t < (int)elems; t += blockDim.x) s_gt[t] = gt[t];
#endif
  __syncthreads();

  const int i = blockIdx.x * blockDim.x + threadIdx.x;
  if (i >= n_rois) return;

  const float4 r  = reinterpret_cast<const float4*>(rois)[i];
  const float  a1 = (r.z - r.x) * (r.w - r.y);

  float best = -3.402823466e38f;
  int   bi   = 0;
#pragma unroll 4
  for (int j = 0; j < ngt; ++j) {
    const float g0 = s_gt[j * 4 + 0];   // LDS broadcast: all lanes same addr
    const float g1 = s_gt[j * 4 + 1];
    const float g2 = s_gt[j * 4 + 2];
    const float g3 = s_gt[j * 4 + 3];
    const float a2 = (g2 - g0) * (g3 - g1);
    const float w  = fmaxf(fminf(r.z, g2) - fmaxf(r.x, g0), 0.0f);
    const float h  = fmaxf(fminf(r.w, g3) - fmaxf(r.y, g1), 0.0f);
    const float inter = w * h;
    const float iou   = inter / (a1 + a2 - inter + 1e-8f);
    if (iou > best) { best = iou; bi = j; }   // first-max == jnp.argmax
  }

  const bool pos = best >= POS_THR;
  const bool neg = (best >= 0.0f) && (best < POS_THR);
  const float rnd = jax_uniform_key42(i, n_rois);

  argm[i]    = bi;
  key_pos[i] = pos ? rnd : 2.0f;
  key_neg[i] = neg ? rnd : 2.0f;

  // Wave-aggregated counting: 1 leader-lane atomic per wave per counter
  // instead of per-thread same-address RMWs (which serialize at L2).
  const u64 act = __ballot(1);
  const u64 bp  = __ballot(pos ? 1 : 0);
  const u64 bn  = __ballot(neg ? 1 : 0);
  const int lane   = (int)(threadIdx.x & 31u);
  const int leader = __ffsll(act) - 1;
  if (lane == leader) {
    const int np = __popcll(bp);
    const int nn = __popcll(bn);
    if (np) atomicAdd(&cnt[0], np);
    if (nn) atomicAdd(&cnt[1], nn);
  }
}

// ---------------------------------------------------------------------------
// Stage 2: stable top-K extraction (== first K entries of jnp.argsort(key)).
// Packed u64 (key_bits<<32 | idx): keys are non-negative floats so uint bit
// order == float order; ties resolve to smaller index (stable-sort semantics).
// ---------------------------------------------------------------------------
__global__ void __launch_bounds__(1024)
pt_topk(float* __restrict__ key, int n, int* __restrict__ sel, int K) {
  __shared__ u64 sm[1024];
  const int tid = threadIdx.x;
  for (int k = 0; k < K; ++k) {
    u64 best = ~0ull;
    for (int i = tid; i < n; i += (int)blockDim.x) {
      const u64 p = ((u64)__float_as_uint(key[i]) << 32) | (u32)i;
      best = umin64(best, p);
    }
    sm[tid] = best;
    __syncthreads();
    for (int s = (int)blockDim.x >> 1; s > 0; s >>= 1) {
      if (tid < s) sm[tid] = umin64(sm[tid], sm[tid + s]);
      __syncthreads();
    }
    if (tid == 0) {
      const u64 w  = sm[0];
      const int ix = (int)(w & 0xFFFFFFFFull);
      sel[k]  = ix;
      key[ix] = __uint_as_float(0x7F800000u);   // +inf: exclude from later rounds
    }
    __syncthreads();
  }
}

// ---------------------------------------------------------------------------
// Stage 3: assemble outputs.
// d_out (float): [0:128) training_idx, [128:256) cls labels,
//                [256:384) reg (32x4), [384] cur_pos.
// ---------------------------------------------------------------------------
__global__ void __launch_bounds__(128)
pt_finalize(const float* __restrict__ rois, const float* __restrict__ gt,
            const int* __restrict__ gt_labels, const int* __restrict__ argm,
            const int* __restrict__ sel_pos, const int* __restrict__ sel_neg,
            const int* __restrict__ cnt, float* __restrict__ out) {
  const int tid = threadIdx.x;
  const int total_pos = cnt[0];
  const int total_neg = cnt[1];
  const int cur_pos = total_pos < MAX_POS ? total_pos : MAX_POS;
  int cur_neg = TOTAL_SAMPLES - cur_pos;
  if (cur_neg > total_neg) cur_neg = total_neg;
  const bool valid = tid < cur_pos + cur_neg;

  int idx = 0;
  if (tid < cur_pos)      idx = sel_pos[tid];
  else if (valid)         idx = sel_neg[tid - cur_pos];
  out[tid] = (float)idx;

  int cls = -1;
  if (valid) cls = (tid < cur_pos) ? gt_labels[argm[idx]] : 0;  // labels==1 / 0
  out[128 + tid] = (float)cls;

  if (tid < MAX_POS) {
    const int p = sel_pos[tid];
    const float4 r  = reinterpret_cast<const float4*>(rois)[p];
    const float4 gb = reinterpret_cast<const float4*>(gt)[argm[p]];
    const float pw0 = r.z - r.x,  ph0 = r.w - r.y;
    const float gw0 = gb.z - gb.x, gh0 = gb.w - gb.y;
    const float pcx = r.x + 0.5f * pw0, pcy = r.y + 0.5f * ph0;  // pre-eps, as reference
    const float gcx = gb.x + 0.5f * gw0, gcy = gb.y + 0.5f * gh0;
    const float pw = fmaxf(pw0, 1e-6f), ph = fmaxf(ph0, 1e-6f);
    const float gw = fmaxf(gw0, 1e-6f), gh = fmaxf(gh0, 1e-6f);
    const bool pv = tid < cur_pos;
    const float t0 = ((gcx - pcx) / pw) / 0.1f;
    const float t1 = ((gcy - pcy) / ph) / 0.1f;
    const float t2 = logf(gw / pw) / 0.2f;
    const float t3 = logf(gh / ph) / 0.2f;
    out[256 + tid * 4 + 0] = pv ? t0 : 0.0f;
    out[256 + tid * 4 + 1] = pv ? t1 : 0.0f;
    out[256 + tid * 4 + 2] = pv ? t2 : 0.0f;
    out[256 + tid * 4 + 3] = pv ? t3 : 0.0f;
  }
  if (tid == 0) out[384] = (float)cur_pos;
}

extern "C" void kernel_launch(void* const* d_in, const int* in_sizes, int n_in,
                              void* d_out, int out_size, void* d_ws, size_t ws_size,
                              hipStream_t stream) {
  const float* rois      = (const float*)d_in[0];
  const float* gt        = (const float*)d_in[1];
  const int*   gt_labels = (const int*)d_in[2];
  // d_in[3] (image_shape) unused by the reference math.

  const int n   = in_sizes[0] / 4;   // 100000 ROIs
  const int ngt = in_sizes[1] / 4;   // 512 GT boxes
  float* out = (float*)d_out;

  // Workspace layout (~1.2 MB).
  float* key_pos = (float*)d_ws;
  float* key_neg = key_pos + n;
  int*   argm    = (int*)(key_neg + n);
  int*   sel_pos = argm + n;
  int*   sel_neg = sel_pos + MAX_POS;
  int*   cnt     = sel_neg + TOTAL_SAMPLES;

  pt_init<<<1, 1, 0, stream>>>(cnt);
  const int blocks = (n + 255) / 256;
  pt_stage1<<<blocks, 256, 0, stream>>>(rois, gt, n, ngt, key_pos, key_neg, argm, cnt);
  pt_topk<<<1, 1024, 0, stream>>>(key_pos, n, sel_pos, MAX_POS);
  pt_topk<<<1, 1024, 0, stream>>>(key_neg, n, sel_neg, TOTAL_SAMPLES);
  pt_finalize<<<1, 128, 0, stream>>>(rois, gt, gt_labels, argm, sel_pos, sel_neg, cnt, out);
}